// DiscrimLoss_18485539242916
// MI455X (gfx1250) — compile-verified
//
#include <hip/hip_runtime.h>
#include <hip/hip_bf16.h>

typedef __attribute__((ext_vector_type(2))) float v2f;
typedef __attribute__((ext_vector_type(8))) float v8f;

// Problem shape (fixed by the reference)
constexpr int BATCH = 8;
constexpr int DCH   = 128;
constexpr int HW    = 65536;            // 256*256
constexpr int NPIX  = BATCH * HW;       // 524288
constexpr int NCLS  = 19;

constexpr float DELTA_V = 0.5f;
constexpr float DELTA_D = 1.5f;
constexpr float ALPHA   = 1.0f;
constexpr float BETA    = 1.0f;
constexpr float GAMMA   = 0.001f;
constexpr float MAXV    = 100.0f;

// Workspace layout (float offsets)
constexpr int WS_SUMS = 0;      // 19*128 = 2432 class sums
constexpr int WS_CNT  = 2432;   // 19 counts (padded to 32)
constexpr int WS_VAR  = 2464;   // 19 var accumulators (padded to 32)
constexpr int WS_MEAN = 2496;   // 19*128 means
constexpr int WS_FLAG = 4928;   // 19 valid-class flags
constexpr int WS_ZERO_LEN = 4960;

// ---------------------------------------------------------------- zero ws
__global__ void zero_ws_kernel(float* __restrict__ ws) {
    int i = blockIdx.x * blockDim.x + threadIdx.x;
    if (i < WS_ZERO_LEN) ws[i] = 0.0f;
}

// ---------------------------------------------------------------- counts
__global__ void count_kernel(const int* __restrict__ labels, float* __restrict__ ws) {
    __shared__ float h[32];
    if (threadIdx.x < 32) h[threadIdx.x] = 0.0f;
    __syncthreads();
    int stride = blockDim.x * gridDim.x;
    for (int i = blockIdx.x * blockDim.x + threadIdx.x; i < NPIX; i += stride) {
        int lab = labels[i];
        if (lab >= 0 && lab < NCLS) atomicAdd(&h[lab], 1.0f);
    }
    __syncthreads();
    if (threadIdx.x < NCLS) atomicAdd(&ws[WS_CNT + threadIdx.x], h[threadIdx.x]);
}

// ------------------------------------------------- segment sums via WMMA
// sums[c][d] += onehot(lab, c) * f[d][pixel]   as D = A(16x4) x B(4x16) + C
// Each wave owns one 16-channel d-tile (wave_id & 7) and a contiguous pixel
// range; two M-tiles cover classes 0..15 and 16..18. The group loop is
// software-pipelined 4x so each wave keeps 8 loads in flight.
constexpr int SS_BLOCKS  = 512;
constexpr int SS_THREADS = 256;                       // 8 waves per block
constexpr int NGROUPS    = NPIX / 4;                  // 131072 groups of 4 px
constexpr int NSEG       = SS_BLOCKS * (SS_THREADS / 32) / 8;  // 512 segments
constexpr int GPS        = NGROUPS / NSEG;            // 256 groups per segment
constexpr int UNROLL     = 4;                          // groups per pipeline step

__global__ void seg_sum_wmma_kernel(const float* __restrict__ feats,
                                    const int* __restrict__ labels,
                                    float* __restrict__ ws) {
    const int lane  = threadIdx.x & 31;
    const int wave  = blockIdx.x * (SS_THREADS >> 5) + (threadIdx.x >> 5);
    const int dtile = wave & 7;          // which 16-channel tile
    const int seg   = wave >> 3;         // pixel segment [0, NSEG)
    const int nidx  = lane & 15;         // N index (channel within tile / class)
    const int half  = lane >> 4;         // upper/lower half-wave

    const int dch = dtile * 16 + nidx;   // this lane's channel (B-matrix N)
    const int m1  = nidx;                // A tile 0: classes 0..15
    const int m2  = 16 + nidx;           // A tile 1: classes 16..31 (only 16..18 hit)

    float* sums = ws + WS_SUMS;

    v8f acc0 = {};
    v8f acc1 = {};

    const int g0 = seg * GPS;
    for (int g = g0; g < g0 + GPS; g += UNROLL) {
        int2   lp[UNROLL];
        float2 fp[UNROLL];

        // Stage 1: issue all loads (8 x global_load_b64 clustered; the waits
        // then count down before each WMMA pair -> deep pipelining).
#pragma unroll
        for (int u = 0; u < UNROLL; ++u) {
            const int p0 = (g + u) * 4;              // 4 pixels per WMMA (K=4)
            const int b  = p0 >> 16;                 // batch
            const int hw = p0 & (HW - 1);
            lp[u] = *reinterpret_cast<const int2*>(labels + p0 + 2 * half);
            const float* fptr =
                feats + ((size_t)(b * DCH + dch) << 16) + hw + 2 * half;
            fp[u] = *reinterpret_cast<const float2*>(fptr);
            if (u == 0) __builtin_prefetch(fptr + 64, 0, 3);  // 256B ahead, WGP scope
        }

        // Stage 2: one-hot A tiles + WMMA accumulation.
#pragma unroll
        for (int u = 0; u < UNROLL; ++u) {
            v2f a0, a1, bv;
            a0.x = (lp[u].x == m1) ? 1.0f : 0.0f;    // K = 0 + 2*half
            a0.y = (lp[u].y == m1) ? 1.0f : 0.0f;    // K = 1 + 2*half
            a1.x = (lp[u].x == m2) ? 1.0f : 0.0f;
            a1.y = (lp[u].y == m2) ? 1.0f : 0.0f;
            bv.x = fp[u].x;
            bv.y = fp[u].y;
            // (neg_a, A, neg_b, B, c_mod, C, reuse_a, reuse_b)
            acc0 = __builtin_amdgcn_wmma_f32_16x16x4_f32(false, a0, false, bv,
                                                         (short)0, acc0, false, false);
            acc1 = __builtin_amdgcn_wmma_f32_16x16x4_f32(false, a1, false, bv,
                                                         (short)0, acc1, false, false);
        }
    }

    // C/D layout: VGPR r, lane l -> (M = r + 8*(l>>4), N = l&15)
#pragma unroll
    for (int r = 0; r < 8; ++r) {
        const int c = r + 8 * half;
        atomicAdd(&sums[c * DCH + dch], acc0[r]);
        const int c2 = 16 + r + 8 * half;
        if (c2 < NCLS) atomicAdd(&sums[c2 * DCH + dch], acc1[r]);
    }
}

// ---------------------------------------------------------------- means
__global__ void means_kernel(float* __restrict__ ws) {
    const int tid = threadIdx.x;
    if (tid < NCLS) ws[WS_FLAG + tid] = (ws[WS_CNT + tid] > MAXV) ? 1.0f : 0.0f;
    for (int i = tid; i < NCLS * DCH; i += blockDim.x) {
        const int c = i >> 7;
        const float safe = fmaxf(ws[WS_CNT + c], 1.0f);
        ws[WS_MEAN + i] = ws[WS_SUMS + i] / safe;
    }
}

// ------------------------------------------------- loss_var pixel pass
constexpr int VP_BLOCKS  = 1024;
constexpr int VP_THREADS = 256;
constexpr int NCHUNK     = NPIX / VP_THREADS;   // 2048, each chunk intra-batch

__global__ void var_pass_kernel(const float* __restrict__ feats,
                                const int* __restrict__ labels,
                                float* __restrict__ ws) {
    __shared__ float smean[DCH * 20];   // transposed + padded: [d*20 + c]
    __shared__ float sflag[32];
    __shared__ float svar[32];

    const float* means = ws + WS_MEAN;
    for (int i = threadIdx.x; i < NCLS * DCH; i += blockDim.x) {
        const int c = i >> 7, d = i & 127;
        smean[d * 20 + c] = means[i];
    }
    if (threadIdx.x < 32) {
        sflag[threadIdx.x] = (threadIdx.x < NCLS) ? ws[WS_FLAG + threadIdx.x] : 0.0f;
        svar[threadIdx.x]  = 0.0f;
    }
    __syncthreads();

    for (int chunk = blockIdx.x; chunk < NCHUNK; chunk += gridDim.x) {
        const int pix = chunk * VP_THREADS + threadIdx.x;
        const int b   = pix >> 16;
        const int hw  = pix & (HW - 1);
        const int lab = labels[pix];
        const bool valid = (lab != -1);
        const int lab0 = valid ? lab : 0;

        float sq = 0.0f;
        const float* fptr = feats + (size_t)b * DCH * HW + hw;
#pragma unroll 8
        for (int d = 0; d < DCH; ++d) {
            const float fv = fptr[(size_t)d * HW];   // coalesced across lanes
            const float df = fv - smean[d * 20 + lab0];
            sq = fmaf(df, df, sq);
        }
        const float dist = (sq > 0.0f) ? sqrtf(sq) : 0.0f;
        const float h = fmaxf(dist - DELTA_V, 0.0f);
        const float contrib = (valid && (sflag[lab0] > 0.5f)) ? h * h : 0.0f;
        atomicAdd(&svar[lab0], contrib);
    }
    __syncthreads();
    if (threadIdx.x < NCLS) atomicAdd(&ws[WS_VAR + threadIdx.x], svar[threadIdx.x]);
}

// ---------------------------------------------------------------- finalize
__global__ void finalize_kernel(float* __restrict__ ws, float* __restrict__ out) {
    __shared__ float sv[NCLS];
    __shared__ int   slast;
    __shared__ float sdist, sreg, svarsum, st;
    const int tid = threadIdx.x;

    if (tid == 0) { sdist = 0.0f; sreg = 0.0f; svarsum = 0.0f; st = 0.0f; slast = -1; }
    if (tid < NCLS) sv[tid] = ws[WS_FLAG + tid];
    __syncthreads();

    if (tid == 0) {
        int lv = -1; float t = 0.0f;
        for (int c = 0; c < NCLS; ++c)
            if (sv[c] > 0.5f) { lv = c; t += 1.0f; }
        slast = lv; st = t;
    }
    __syncthreads();

    // loss_dist: a over all valid classes; b over valid classes except the
    // last valid one (faithful to reference, includes a == b diagonal).
    if (tid < NCLS * NCLS) {
        const int a = tid / NCLS, bc = tid % NCLS;
        const bool bm = (sv[bc] > 0.5f) && (bc != slast);
        if ((sv[a] > 0.5f) && bm) {
            float sq = 0.0f;
            const float* ma = ws + WS_MEAN + a * DCH;
            const float* mb = ws + WS_MEAN + bc * DCH;
            for (int d = 0; d < DCH; ++d) {
                const float df = ma[d] - mb[d];
                sq = fmaf(df, df, sq);
            }
            const float pdn = (sq > 0.0f) ? sqrtf(sq) : 0.0f;
            const float hd = fmaxf(2.0f * DELTA_D - pdn, 0.0f);
            atomicAdd(&sdist, hd * hd);
        }
    }
    // loss_reg + loss_var
    if (tid < NCLS && sv[tid] > 0.5f) {
        float sq = 0.0f;
        const float* m = ws + WS_MEAN + tid * DCH;
        for (int d = 0; d < DCH; ++d) sq = fmaf(m[d], m[d], sq);
        const float mn = (sq > 0.0f) ? sqrtf(sq) : 0.0f;
        atomicAdd(&sreg, mn);
        atomicAdd(&svarsum, ws[WS_VAR + tid] / fmaxf(ws[WS_CNT + tid], 1.0f));
    }
    __syncthreads();

    if (tid == 0) {
        const float t = st;
        out[0] = ALPHA * svarsum / t
               + BETA  * sdist / (t * (t - 1.0f))
               + GAMMA * sreg / t;
    }
}

// ---------------------------------------------------------------- launch
extern "C" void kernel_launch(void* const* d_in, const int* in_sizes, int n_in,
                              void* d_out, int out_size, void* d_ws, size_t ws_size,
                              hipStream_t stream) {
    (void)in_sizes; (void)n_in; (void)out_size; (void)ws_size;
    const float* feats  = (const float*)d_in[0];
    const int*   labels = (const int*)d_in[1];
    float* out = (float*)d_out;
    float* ws  = (float*)d_ws;

    hipLaunchKernelGGL(zero_ws_kernel, dim3((WS_ZERO_LEN + 255) / 256), dim3(256), 0, stream, ws);
    hipLaunchKernelGGL(count_kernel, dim3(256), dim3(256), 0, stream, labels, ws);
    hipLaunchKernelGGL(seg_sum_wmma_kernel, dim3(SS_BLOCKS), dim3(SS_THREADS), 0, stream,
                       feats, labels, ws);
    hipLaunchKernelGGL(means_kernel, dim3(1), dim3(256), 0, stream, ws);
    hipLaunchKernelGGL(var_pass_kernel, dim3(VP_BLOCKS), dim3(VP_THREADS), 0, stream,
                       feats, labels, ws);
    hipLaunchKernelGGL(finalize_kernel, dim3(1), dim3(384), 0, stream, ws, out);
}